// QuantileGCN_63015760166988
// MI455X (gfx1250) — compile-verified
//
#include <hip/hip_runtime.h>
#include <hip/hip_bf16.h>

// ---------------------------------------------------------------------------
// QuantileGCN on MI455X (gfx1250): 2-layer GCN, fp32 WMMA for the dense GEMMs,
// L2-resident atomics for the edge scatter (output arrays fit in 192MB L2).
// ---------------------------------------------------------------------------

typedef __attribute__((ext_vector_type(2))) float v2f;
typedef __attribute__((ext_vector_type(8))) float v8f;

#define DIN 64
#define NQ 3

// --- K0: init deg=1 (self loop), zero out1 and final out ---------------------
__global__ void k_init(float* __restrict__ deg, float* __restrict__ out1,
                       float* __restrict__ out, int n_nodes) {
  int idx = blockIdx.x * blockDim.x + threadIdx.x;
  int tot = n_nodes * DIN;
  if (idx < tot) out1[idx] = 0.0f;
  if (idx < n_nodes) deg[idx] = 1.0f;
  if (idx < n_nodes * NQ) out[idx] = 0.0f;
}

// --- K1: degree accumulation over edges (dst) --------------------------------
__global__ void k_deg(const int* __restrict__ ei, float* __restrict__ deg, int E) {
  int e = blockIdx.x * blockDim.x + threadIdx.x;
  if (e >= E) return;
  int d = ei[E + e];
  atomicAdd(&deg[d], 1.0f);
}

// --- K2: dis = rsqrt(deg)  (deg >= 1 always due to self loops) ---------------
__global__ void k_dis(float* __restrict__ deg, int n_nodes) {
  int i = blockIdx.x * blockDim.x + threadIdx.x;
  if (i >= n_nodes) return;
  deg[i] = __frsqrt_rn(deg[i]);
}

// --- K3: h = x @ W1 via V_WMMA_F32_16X16X4_F32 -------------------------------
// Wave computes a 16-row x 64-col strip. W1 staged transposed in LDS so the
// B fragment (two consecutive K values for one column) is a contiguous b64.
__global__ void k_gemm1(const float* __restrict__ x, const float* __restrict__ W,
                        float* __restrict__ h, int n_rows) {
  __shared__ float lwT[DIN * DIN]; // lwT[n*64 + k] = W[k*64 + n] (16KB)
  for (int i = threadIdx.x; i < DIN * DIN; i += blockDim.x) {
    int k = i / DIN, n = i % DIN;
    lwT[n * DIN + k] = W[i];
  }
  __syncthreads();

  int wave = threadIdx.x >> 5;
  int lane = threadIdx.x & 31;
  int tile = blockIdx.x * (blockDim.x >> 5) + wave; // 16-row tile
  if (tile * 16 >= n_rows) return;                  // wave-uniform: EXEC stays full

  int row0 = tile * 16;
  int laneM = lane & 15;
  int kOff = (lane >> 4) * 2; // lanes 0-15: K=k0,k0+1 ; lanes 16-31: K=k0+2,k0+3

  v8f acc0 = {}, acc1 = {}, acc2 = {}, acc3 = {};
  const float* xrow = x + (size_t)(row0 + laneM) * DIN;

#pragma unroll
  for (int k0 = 0; k0 < DIN; k0 += 4) {
    v2f a = *(const v2f*)(xrow + k0 + kOff);
    v2f b0 = *(const v2f*)(lwT + (size_t)(0 * 16 + laneM) * DIN + k0 + kOff);
    v2f b1 = *(const v2f*)(lwT + (size_t)(1 * 16 + laneM) * DIN + k0 + kOff);
    v2f b2 = *(const v2f*)(lwT + (size_t)(2 * 16 + laneM) * DIN + k0 + kOff);
    v2f b3 = *(const v2f*)(lwT + (size_t)(3 * 16 + laneM) * DIN + k0 + kOff);
    acc0 = __builtin_amdgcn_wmma_f32_16x16x4_f32(false, a, false, b0, (short)0, acc0, false, false);
    acc1 = __builtin_amdgcn_wmma_f32_16x16x4_f32(false, a, false, b1, (short)0, acc1, false, false);
    acc2 = __builtin_amdgcn_wmma_f32_16x16x4_f32(false, a, false, b2, (short)0, acc2, false, false);
    acc3 = __builtin_amdgcn_wmma_f32_16x16x4_f32(false, a, false, b3, (short)0, acc3, false, false);
  }

  int hi8 = (lane >> 4) * 8; // lanes 16-31 hold rows r+8
#pragma unroll
  for (int r = 0; r < 8; ++r) {
    size_t row = (size_t)(row0 + r + hi8) * DIN;
    h[row + 0 * 16 + laneM] = acc0[r];
    h[row + 1 * 16 + laneM] = acc1[r];
    h[row + 2 * 16 + laneM] = acc2[r];
    h[row + 3 * 16 + laneM] = acc3[r];
  }
}

// --- K4: edge scatter layer 1: out1[dst] += norm * h[src] --------------------
// One wave per edge: 64 channels -> float2 per lane, 2 f32 atomics (L2-resident).
__global__ void k_edge1(const int* __restrict__ ei, const float* __restrict__ dis,
                        const float* __restrict__ h, float* __restrict__ out1, int E) {
  int warpId = (blockIdx.x * blockDim.x + threadIdx.x) >> 5;
  int lane = threadIdx.x & 31;
  if (warpId >= E) return;
  int s = ei[warpId];
  int d = ei[E + warpId];
  float norm = dis[s] * dis[d];
  const float2 hv = *(const float2*)(h + (size_t)s * DIN + lane * 2);
  float* o = out1 + (size_t)d * DIN + lane * 2;
  atomicAdd(o + 0, norm * hv.x);
  atomicAdd(o + 1, norm * hv.y);
}

// --- K5: finalize layer 1: out1 += dis^2 * h (self loop) + b1 ----------------
__global__ void k_fin1(const float* __restrict__ dis, const float* __restrict__ h,
                       const float* __restrict__ b1, float* __restrict__ out1, int n_nodes) {
  int idx = blockIdx.x * blockDim.x + threadIdx.x;
  if (idx >= n_nodes * DIN) return;
  int i = idx / DIN, c = idx % DIN;
  float di = dis[i];
  out1[idx] += di * di * h[idx] + b1[c];
}

// --- K6: h2 = out1 @ W2 (64x3, padded to 16 cols) via WMMA -------------------
__global__ void k_gemm2(const float* __restrict__ out1, const float* __restrict__ W2,
                        float* __restrict__ h2, int n_rows) {
  __shared__ float lwT[16 * DIN]; // lwT[n*64 + k] = (n<3) ? W2[k*3+n] : 0 (4KB)
  for (int i = threadIdx.x; i < 16 * DIN; i += blockDim.x) {
    int n = i / DIN, k = i % DIN;
    lwT[i] = (n < NQ) ? W2[k * NQ + n] : 0.0f;
  }
  __syncthreads();

  int wave = threadIdx.x >> 5;
  int lane = threadIdx.x & 31;
  int tile = blockIdx.x * (blockDim.x >> 5) + wave;
  if (tile * 16 >= n_rows) return; // wave-uniform

  int row0 = tile * 16;
  int laneM = lane & 15;
  int kOff = (lane >> 4) * 2;

  v8f acc = {};
  const float* arow = out1 + (size_t)(row0 + laneM) * DIN;
#pragma unroll
  for (int k0 = 0; k0 < DIN; k0 += 4) {
    v2f a = *(const v2f*)(arow + k0 + kOff);
    v2f b = *(const v2f*)(lwT + (size_t)laneM * DIN + k0 + kOff);
    acc = __builtin_amdgcn_wmma_f32_16x16x4_f32(false, a, false, b, (short)0, acc, false, false);
  }

  int hi8 = (lane >> 4) * 8;
  if (laneM < NQ) {
#pragma unroll
    for (int r = 0; r < 8; ++r) {
      h2[(size_t)(row0 + r + hi8) * NQ + laneM] = acc[r];
    }
  }
}

// --- K7: edge scatter layer 2: out[dst] += norm * h2[src] (3 ch) -------------
__global__ void k_edge2(const int* __restrict__ ei, const float* __restrict__ dis,
                        const float* __restrict__ h2, float* __restrict__ out, int E) {
  int e = blockIdx.x * blockDim.x + threadIdx.x;
  if (e >= E) return;
  int s = ei[e];
  int d = ei[E + e];
  float norm = dis[s] * dis[d];
  const float* hs = h2 + (size_t)s * NQ;
  float* o = out + (size_t)d * NQ;
#pragma unroll
  for (int c = 0; c < NQ; ++c) atomicAdd(o + c, norm * hs[c]);
}

// --- K8: finalize layer 2: out += dis^2 * h2 + b2 ----------------------------
__global__ void k_fin2(const float* __restrict__ dis, const float* __restrict__ h2,
                       const float* __restrict__ b2, float* __restrict__ out, int n_nodes) {
  int idx = blockIdx.x * blockDim.x + threadIdx.x;
  if (idx >= n_nodes * NQ) return;
  int i = idx / NQ, c = idx % NQ;
  float di = dis[i];
  out[idx] += di * di * h2[idx] + b2[c];
}

// ---------------------------------------------------------------------------
extern "C" void kernel_launch(void* const* d_in, const int* in_sizes, int n_in,
                              void* d_out, int out_size, void* d_ws, size_t ws_size,
                              hipStream_t stream) {
  const float* x  = (const float*)d_in[0];
  const int*   ei = (const int*)d_in[1]; // [2, E] src row then dst row
  const float* W1 = (const float*)d_in[2];
  const float* b1 = (const float*)d_in[3];
  const float* W2 = (const float*)d_in[4];
  const float* b2 = (const float*)d_in[5];
  float* out = (float*)d_out;

  const int N = in_sizes[0] / DIN;  // 100000
  const int E = in_sizes[1] / 2;    // 1200000

  // Workspace layout (floats): dis[N] | h[N*64] | out1[N*64] | h2[N*3]
  float* ws  = (float*)d_ws;
  float* dis = ws;
  float* h   = ws + N;
  float* out1= ws + N + (size_t)N * DIN;
  float* h2  = ws + N + (size_t)2 * N * DIN;

  const int B = 256;

  // K0: init
  {
    int tot = N * DIN;
    k_init<<<(tot + B - 1) / B, B, 0, stream>>>(dis, out1, out, N);
  }
  // K1: degree
  k_deg<<<(E + B - 1) / B, B, 0, stream>>>(ei, dis, E);
  // K2: dis = rsqrt(deg)
  k_dis<<<(N + B - 1) / B, B, 0, stream>>>(dis, N);
  // K3: h = x @ W1 (WMMA)
  {
    int tiles = (N + 15) / 16;               // 6250
    int wavesPerBlock = B / 32;              // 8
    int blocks = (tiles + wavesPerBlock - 1) / wavesPerBlock;
    k_gemm1<<<blocks, B, 0, stream>>>(x, W1, h, N);
  }
  // K4: edge scatter layer 1 (wave per edge)
  {
    long long threads = (long long)E * 32;
    int blocks = (int)((threads + B - 1) / B);
    k_edge1<<<blocks, B, 0, stream>>>(ei, dis, h, out1, E);
  }
  // K5: finalize 1 (self loop + bias)
  k_fin1<<<(N * DIN + B - 1) / B, B, 0, stream>>>(dis, h, b1, out1, N);
  // K6: h2 = out1 @ W2 (WMMA)
  {
    int tiles = (N + 15) / 16;
    int wavesPerBlock = B / 32;
    int blocks = (tiles + wavesPerBlock - 1) / wavesPerBlock;
    k_gemm2<<<blocks, B, 0, stream>>>(out1, W2, h2, N);
  }
  // K7: edge scatter layer 2 (thread per edge)
  k_edge2<<<(E + B - 1) / B, B, 0, stream>>>(ei, dis, h2, out, E);
  // K8: finalize 2
  k_fin2<<<(N * NQ + B - 1) / B, B, 0, stream>>>(dis, h2, b2, out, N);
}